// DepthwiseConv_5042291605794
// MI455X (gfx1250) — compile-verified
//
#include <hip/hip_runtime.h>

typedef __attribute__((ext_vector_type(2))) float v2f;
typedef __attribute__((ext_vector_type(4))) float v4f;
typedef __attribute__((ext_vector_type(8))) float v8f;

#define D_IN   64
#define D_RAD  128
#define ROWP   132          // padded LDS row stride (floats): conflict-free b64 reads
#define WAVES  8
#define TILE_E 16

// Zero-initialize the output accumulator (harness poisons d_out with 0xAA).
__global__ void __launch_bounds__(256)
zero_out_kernel(float* __restrict__ out, int n)
{
    int i = blockIdx.x * 256 + threadIdx.x;
    if (i < n) out[i] = 0.0f;
}

// Fused: filt = edge_basis @ W^T + b  (fp32 WMMA 16x16x4),
//        m = x[src] * filt, h[dst] += m (global_atomic_add_f32).
// One wave per 16-edge tile; 8 waves (256 threads) per block.
__global__ void __launch_bounds__(WAVES * 32)
egnn_radial_gather_scatter(const float* __restrict__ x,
                           const float* __restrict__ eb,
                           const int*   __restrict__ src,
                           const int*   __restrict__ dst,
                           const float* __restrict__ W,
                           const float* __restrict__ bias,
                           float*       __restrict__ out,
                           int n_edges, int n_tiles)
{
    __shared__ float sW[D_IN * ROWP];            // 33,792 B, shared by all waves
    __shared__ float sA[WAVES][TILE_E * ROWP];   // 8,448 B per wave

    const int tid  = threadIdx.x;
    const int wave = tid >> 5;
    const int lane = tid & 31;
    const int n    = lane & 15;   // N-column within 16-wide tile
    const int hi   = lane >> 4;   // upper half-wave flag

    // ---- stage W (64x128 f32) into padded LDS, coalesced float4 ----
#pragma unroll
    for (int i = 0; i < (D_IN * D_RAD) / (WAVES * 32 * 4); ++i) {
        int f   = (tid + i * (WAVES * 32)) * 4;   // flat float index (mult of 4)
        int row = f >> 7;                          // / 128
        int col = f & (D_RAD - 1);
        *(v4f*)(&sW[row * ROWP + col]) = *(const v4f*)(W + f);
    }

    const int  tile   = blockIdx.x * WAVES + wave;
    const bool active = (tile < n_tiles);          // wave-uniform: EXEC stays full
    const int  ebase  = tile * TILE_E;

    if (active) {
        // 16x128 edge_basis tile is contiguous: fully coalesced b128 loads
        const float* tp = eb + (long long)ebase * D_RAD;
#pragma unroll
        for (int j = 0; j < TILE_E; ++j) {
            *(v4f*)(&sA[wave][j * ROWP + lane * 4]) =
                *(const v4f*)(tp + j * D_RAD + lane * 4);
        }
    }
    __syncthreads();

    if (active) {
        // A (16x4 f32): lane L<16 holds M=L, {K0,K1}; lane 16+L holds {K2,K3}
        // B (4x16 f32): lane L<16 holds {B[0][L],B[1][L]}; lane 16+L {B[2][L],B[3][L]}
        //   with B[k][d] = W[d][k]  -> read W rows d = 16t + n
        const float* aw = &sA[wave][n * ROWP + 2 * hi];
        const float* w0 = &sW[(n +  0) * ROWP + 2 * hi];
        const float* w1 = &sW[(n + 16) * ROWP + 2 * hi];
        const float* w2 = &sW[(n + 32) * ROWP + 2 * hi];
        const float* w3 = &sW[(n + 48) * ROWP + 2 * hi];

        v8f c0 = {}, c1 = {}, c2 = {}, c3 = {};
#pragma unroll
        for (int k = 0; k < D_RAD; k += 4) {
            v2f a  = *(const v2f*)(aw + k);
            v2f b0 = *(const v2f*)(w0 + k);
            v2f b1 = *(const v2f*)(w1 + k);
            v2f b2 = *(const v2f*)(w2 + k);
            v2f b3 = *(const v2f*)(w3 + k);
            c0 = __builtin_amdgcn_wmma_f32_16x16x4_f32(false, a, false, b0, (short)0, c0, false, false);
            c1 = __builtin_amdgcn_wmma_f32_16x16x4_f32(false, a, false, b1, (short)0, c1, false, false);
            c2 = __builtin_amdgcn_wmma_f32_16x16x4_f32(false, a, false, b2, (short)0, c2, false, false);
            c3 = __builtin_amdgcn_wmma_f32_16x16x4_f32(false, a, false, b3, (short)0, c3, false, false);
        }

        // + bias (per output channel d = 16t + n, identical across M rows)
        float bb0 = bias[n];      float bb1 = bias[n + 16];
        float bb2 = bias[n + 32]; float bb3 = bias[n + 48];
        c0 += bb0; c1 += bb1; c2 += bb2; c3 += bb3;

        // C/D layout: VGPR i -> (M = i or 8+i, N = n). Gather x[src], scatter-add h[dst].
#pragma unroll
        for (int i = 0; i < 8; ++i) {
            int e = ebase + (hi << 3) + i;
            if (e < n_edges) {
                int s = src[e];
                int d = dst[e];
                const float* xs = x   + (long long)s * D_IN;
                float*       hd = out + (long long)d * D_IN;
                atomicAdd(&hd[n +  0], c0[i] * xs[n +  0]);
                atomicAdd(&hd[n + 16], c1[i] * xs[n + 16]);
                atomicAdd(&hd[n + 32], c2[i] * xs[n + 32]);
                atomicAdd(&hd[n + 48], c3[i] * xs[n + 48]);
            }
        }
    }
}

extern "C" void kernel_launch(void* const* d_in, const int* in_sizes, int n_in,
                              void* d_out, int out_size, void* d_ws, size_t ws_size,
                              hipStream_t stream) {
    const float* x    = (const float*)d_in[0];   // [N, 64]
    const float* eb   = (const float*)d_in[1];   // [E, 128]
    const int*   src  = (const int*)  d_in[2];   // [E]
    const int*   dst  = (const int*)  d_in[3];   // [E]
    const float* W    = (const float*)d_in[4];   // [64, 128]
    const float* bias = (const float*)d_in[5];   // [64]
    float* out = (float*)d_out;                  // [N, 64]

    const int E       = in_sizes[2];
    const int n_tiles = (E + TILE_E - 1) / TILE_E;

    // 1) zero the accumulator
    zero_out_kernel<<<(out_size + 255) / 256, 256, 0, stream>>>(out, out_size);

    // 2) fused WMMA GEMM + gather + scatter-add
    const int blocks = (n_tiles + WAVES - 1) / WAVES;
    egnn_radial_gather_scatter<<<blocks, WAVES * 32, 0, stream>>>(
        x, eb, src, dst, W, bias, out, E, n_tiles);
}